// optimizer_72215580115445
// MI455X (gfx1250) — compile-verified
//
#include <hip/hip_runtime.h>
#include <math.h>

// dual-ascent optimizer, LDS-resident per (W,B) pair, fp32 WMMA 16x16x4 matmuls
// v3: batch-load Omega B-fragments into distinct VGPRs (pipeline DS latency),
//     dual accumulator chains in Phase A.
#define MU 1e-3f
#define RO 1e-3f

typedef float v2f __attribute__((ext_vector_type(2)));
typedef float v8f __attribute__((ext_vector_type(8)));

constexpr int S = 16;
constexpr int G = 1024;
constexpr int NW = 8;     // 8 waves of 32 (wave32)

__global__ __launch_bounds__(256) void dual_ascent_kernel(
    const float* __restrict__ Yp,    // [WB, S, G]
    const float* __restrict__ Uk,    // [WB, S, S]
    const float* __restrict__ Lk,    // [WB, S, S]
    const float* __restrict__ Om0,   // [WB, G, S]
    const int*   __restrict__ nit,   // scalar
    float* __restrict__ Out)         // [WB, G, S]
{
  __shared__ float s_yp[S * G];        // 64 KB   Yp[s][g]
  __shared__ float s_om[G * S];        // 64 KB   Omega[g][s]
  __shared__ float s_scr[NW * 256];    //  8 KB   per-wave grad_lam partials
  __shared__ float s_inv[G];           //  4 KB   1/||Omega_row||
  __shared__ float s_lam[256];         //  1 KB   lam[i][j]
  __shared__ float s_tgt[256];         //  1 KB   target = Uk @ Lambda_k

  const int tid  = threadIdx.x;
  const int lane = tid & 31;
  const int wv   = tid >> 5;
  const int pair = blockIdx.x;
  const long base_sg = (long)pair * (S * G);

  // ---- stage Yp and Omega0 into LDS (vectorized b128) ----
  {
    const float4* gy = (const float4*)(Yp  + base_sg);
    const float4* go = (const float4*)(Om0 + base_sg);
    float4* ly = (float4*)s_yp;
    float4* lo = (float4*)s_om;
    for (int i = tid; i < (S * G) / 4; i += 256) { ly[i] = gy[i]; lo[i] = go[i]; }
  }
  // ---- target = Uk @ Lambda_k (16x16, one element per thread); lam = 0 ----
  {
    const int i = tid >> 4, j = tid & 15;
    const float* u = Uk + (long)pair * 256;
    const float* l = Lk + (long)pair * 256;
    float acc = 0.f;
    #pragma unroll
    for (int k = 0; k < 16; ++k) acc += u[i * 16 + k] * l[k * 16 + j];
    s_tgt[tid] = acc;
    s_lam[tid] = 0.f;
  }
  __syncthreads();

  const int niters = nit[0];
  const int m      = lane & 15;          // row (A) / col (B,C,D) within tile
  const int khalf  = (lane >> 4) * 2;    // 0 or 2: K-pair selector for A/B frags
  const int chalf  = 8 * (lane >> 4);    // C/D row offset for upper half-wave

  // ---- cache loop-invariant Yp fragments in VGPRs ----
  // Phase A A-fragments: A[m][k] = Yp[m][k0+k], k-chunk owned by this wave
  v2f fA[32];
  #pragma unroll
  for (int kt = 0; kt < 32; ++kt) {
    const int k0 = wv * 128 + kt * 4;
    fA[kt].x = s_yp[m * G + k0 + khalf];
    fA[kt].y = s_yp[m * G + k0 + khalf + 1];
  }
  // Phase B2 A-fragments: A[m][k] = YpT[g0+m][k] = Yp[k][g0+m], 8 G-tiles
  v2f fB[8][4];
  #pragma unroll
  for (int t = 0; t < 8; ++t) {
    const int g0 = wv * 128 + t * 16;
    #pragma unroll
    for (int kk = 0; kk < 4; ++kk) {
      const int k0 = kk * 4;
      fB[t][kk].x = s_yp[(k0 + khalf) * G + g0 + m];
      fB[t][kk].y = s_yp[(k0 + khalf + 1) * G + g0 + m];
    }
  }

  for (int it = 0; it < niters; ++it) {
    // ---- Phase A: partial grad_lam = Yp[:,chunk] @ Omega[chunk,:]  (32 WMMAs) ----
    {
      const int kchunk = wv * 128;
      // batch-load all 32 Omega B-fragments: distinct regs -> DS loads pipeline
      v2f bf[32];
      #pragma unroll
      for (int kt = 0; kt < 32; ++kt) {
        const int k0 = kchunk + kt * 4;
        bf[kt].x = s_om[(k0 + khalf) * S + m];        // B[k][n] = Omega[k][n]
        bf[kt].y = s_om[(k0 + khalf + 1) * S + m];
      }
      // two independent accumulator chains to overlap WMMA accumulate latency
      v8f c0 = {}, c1 = {};
      #pragma unroll
      for (int kt = 0; kt < 32; kt += 2) {
        c0 = __builtin_amdgcn_wmma_f32_16x16x4_f32(false, fA[kt],     false, bf[kt],
                                                   (short)0, c0, false, false);
        c1 = __builtin_amdgcn_wmma_f32_16x16x4_f32(false, fA[kt + 1], false, bf[kt + 1],
                                                   (short)0, c1, false, false);
      }
      #pragma unroll
      for (int r = 0; r < 8; ++r)
        s_scr[wv * 256 + (r + chalf) * 16 + m] = c0[r] + c1[r];
    }
    __syncthreads();

    // ---- Phase B1: inverse row norms of OLD Omega (4 rows per thread) ----
    for (int rr = 0; rr < 4; ++rr) {
      const int g = tid + 256 * rr;
      float acc = 0.f;
      #pragma unroll
      for (int s = 0; s < 16; ++s) { const float v = s_om[g * 16 + s]; acc += v * v; }
      s_inv[g] = 1.0f / sqrtf(acc);
    }
    __syncthreads();

    // ---- Phase B2: Omega update for own 128 rows, using OLD lam ----
    {
      // lam B-fragments depend only on kk: load once, reuse across all 8 tiles
      v2f bl[4];
      #pragma unroll
      for (int kk = 0; kk < 4; ++kk) {
        const int k0 = kk * 4;
        bl[kk].x = s_lam[(k0 + khalf) * S + m];
        bl[kk].y = s_lam[(k0 + khalf + 1) * S + m];
      }
      #pragma unroll
      for (int t = 0; t < 8; ++t) {
        const int g0 = wv * 128 + t * 16;
        v8f d = {};                              // d = YpT_tile @ lam
        #pragma unroll
        for (int kk = 0; kk < 4; ++kk) {
          d = __builtin_amdgcn_wmma_f32_16x16x4_f32(false, fB[t][kk], false, bl[kk],
                                                    (short)0, d, false, false);
        }
        #pragma unroll
        for (int r = 0; r < 8; ++r) {
          const int g = g0 + r + chalf;
          const float om = s_om[g * 16 + m];
          s_om[g * 16 + m] = om - MU * (om * s_inv[g] + d[r]);
        }
      }
    }
    __syncthreads();

    // ---- Phase C: lam += RO * (sum_w partial_w - target) ----
    {
      float acc = 0.f;
      #pragma unroll
      for (int w = 0; w < NW; ++w) acc += s_scr[w * 256 + tid];
      s_lam[tid] += RO * (acc - s_tgt[tid]);
    }
    __syncthreads();
  }

  // ---- write Omega back (layout already matches [G,S] row-major) ----
  {
    const float4* lo = (const float4*)s_om;
    float4* go = (float4*)(Out + base_sg);
    for (int i = tid; i < (S * G) / 4; i += 256) go[i] = lo[i];
  }
}

extern "C" void kernel_launch(void* const* d_in, const int* in_sizes, int n_in,
                              void* d_out, int out_size, void* d_ws, size_t ws_size,
                              hipStream_t stream) {
  const float* Yp  = (const float*)d_in[0];
  const float* Uk  = (const float*)d_in[1];
  const float* Lk  = (const float*)d_in[2];
  const float* Om0 = (const float*)d_in[3];
  const int*   nit = (const int*)d_in[4];
  float* out = (float*)d_out;
  const int WB = out_size / (G * S);   // 4*32 = 128 workgroups, one per (w,b)
  dual_ascent_kernel<<<WB, 256, 0, stream>>>(Yp, Uk, Lk, Om0, nit, out);
}